// GNNLayer_30167850287107
// MI455X (gfx1250) — compile-verified
//
#include <hip/hip_runtime.h>
#include <math.h>

typedef __attribute__((ext_vector_type(2))) float v2f;
typedef __attribute__((ext_vector_type(8))) float v8f;

#define LEAKY(x, s) ((x) > 0.0f ? (x) : (s) * (x))

// Problem constants (from reference setup_inputs)
#define NB 8          // batch
#define NS 16         // sections
#define NN 256        // nodes per section
#define NF 128        // in features
#define NC 128        // out channels
#define NNODE 4096    // S*n nodes per batch (bend graph)
#define NTOT 32768    // B*S*n total node rows
#define EB 65536      // bend edges per batch
#define ES 4096       // section edges per graph
#define NG 128        // B*S section graphs

__device__ __forceinline__ int ordf(float f) {
    int i = __float_as_int(f);
    return i >= 0 ? i : (i ^ 0x7FFFFFFF);
}
__device__ __forceinline__ float unordf(int s) {
    return __int_as_float(s >= 0 ? s : (s ^ 0x7FFFFFFF));
}

// CDNA5 async global->LDS copy (ASYNCcnt-tracked, CDNA5 ISA).
// vdst = LDS byte address (low 32 bits of generic pointer), vaddr = 64-bit global.
__device__ __forceinline__ void async_load_b128(unsigned lds_addr, const float* gaddr) {
    asm volatile("global_load_async_to_lds_b128 %0, %1, off"
                 :: "v"(lds_addr), "v"(gaddr)
                 : "memory");
}
__device__ __forceinline__ void wait_asynccnt0() {
    asm volatile("s_wait_asynccnt 0x0" ::: "memory");
}

// ---------------------------------------------------------------------------
// 0) Fold small linear algebra:
//    params[0..3]  = enc_W @ (We_b @ a_edge_b), params[4] = enc_b . (We_b@a_edge_b)
//    params[5..8]  = enc_W @ (We_s @ a_edge_s), params[9] = enc_b . (We_s@a_edge_s)
//    params[10..11] = softmax(fuse_w)
// ---------------------------------------------------------------------------
__global__ void k_precompute(const float* __restrict__ encW, const float* __restrict__ encB,
                             const float* __restrict__ Web, const float* __restrict__ aeb,
                             const float* __restrict__ Wes, const float* __restrict__ aes,
                             const float* __restrict__ fusew, float* __restrict__ params) {
    if (threadIdx.x != 0 || blockIdx.x != 0) return;
    float web[4], wes[4];
    for (int d = 0; d < 4; ++d) {
        float s1 = 0.f, s2 = 0.f;
        for (int c = 0; c < NC; ++c) {
            s1 += Web[d * NC + c] * aeb[c];
            s2 += Wes[d * NC + c] * aes[c];
        }
        web[d] = s1; wes[d] = s2;
    }
    for (int d = 0; d < 4; ++d) {
        float m1 = 0.f, m2 = 0.f;
        for (int dp = 0; dp < 4; ++dp) {
            m1 += encW[d * 4 + dp] * web[dp];
            m2 += encW[d * 4 + dp] * wes[dp];
        }
        params[d] = m1; params[5 + d] = m2;
    }
    float c1 = 0.f, c2 = 0.f;
    for (int dp = 0; dp < 4; ++dp) { c1 += encB[dp] * web[dp]; c2 += encB[dp] * wes[dp]; }
    params[4] = c1; params[9] = c2;
    float e0 = expf(fusew[0]), e1 = expf(fusew[1]);
    params[10] = e0 / (e0 + e1);
    params[11] = e1 / (e0 + e1);
}

// ---------------------------------------------------------------------------
// 1) WMMA projection GEMM: Hb = X @ Wb, Hs = X @ Ws  (X: [NTOT,128] fp32)
//    Block = 4 waves, fixed ntile. Weight slice [128 x 16] of BOTH matrices is
//    async-copied to LDS once (16 KB), then each wave register-blocks 4 M-tiles:
//    per k-step 4 x b64 A loads + LDS B fragments feed 8 v_wmma_f32_16x16x4_f32.
//    A frag: lanes 0-15 row=lane, K=k..k+1 ; lanes 16-31 K=k+2..k+3
//    B frag: lanes 0-15 col=lane, K=k..k+1 ; lanes 16-31 K=k+2..k+3
//    D: VGPR v -> row = mtile*16 + v + 8*(lane>>4), col = lane&15
// ---------------------------------------------------------------------------
__global__ void __launch_bounds__(128) k_gemm_proj(const float* __restrict__ X,
                                                   const float* __restrict__ Wb,
                                                   const float* __restrict__ Ws,
                                                   float* __restrict__ Hb,
                                                   float* __restrict__ Hs) {
    __shared__ float wtile[2][NF][16];   // [matrix][k][col]  = 16 KB

    const int lane = threadIdx.x;                        // 0..31
    const int ntile = blockIdx.y;                        // 0..7
    const int tid = threadIdx.y * 32 + lane;             // 0..127

    // Stage W slices (cols ntile*16..+16, all k, both matrices) into LDS.
    // 1024 float4 transfers over 128 threads = 8 async b128 loads each.
#pragma unroll
    for (int it = 0; it < 8; ++it) {
        int q = it * 128 + tid;          // float4 index 0..1023
        int m = q >> 9;                  // 0..1 (matrix)
        int rowk = (q >> 2) & 127;       // k row
        int c4 = q & 3;                  // float4 within row
        const float* g = (m ? Ws : Wb) + rowk * NC + ntile * 16 + c4 * 4;
        unsigned lds = (unsigned)(uintptr_t)&wtile[m][rowk][c4 * 4];
        async_load_b128(lds, g);
    }
    wait_asynccnt0();
    __syncthreads();

    const int mgroup = blockIdx.x * blockDim.y + threadIdx.y;  // 0..511
    const int m0 = mgroup * 64;                                // 4 M-tiles
    const int r = lane & 15;
    const int kh = (lane >> 4) * 2;                            // 0 or 2
    const int col = ntile * 16 + r;

    v8f acc[8];
#pragma unroll
    for (int i = 0; i < 8; ++i) acc[i] = (v8f){0.f, 0.f, 0.f, 0.f, 0.f, 0.f, 0.f, 0.f};

    const float* xr0 = X + (size_t)(m0 + r) * NF;
    for (int k = 0; k < NF; k += 4) {
        v2f bB, bS;
        bB.x = wtile[0][k + kh][r];
        bB.y = wtile[0][k + kh + 1][r];
        bS.x = wtile[1][k + kh][r];
        bS.y = wtile[1][k + kh + 1][r];
        v2f a[4];
#pragma unroll
        for (int t = 0; t < 4; ++t) {
            const float* xr = xr0 + (size_t)(t * 16) * NF;
            a[t].x = xr[k + kh];
            a[t].y = xr[k + kh + 1];
        }
#pragma unroll
        for (int t = 0; t < 4; ++t) {
            acc[2 * t] = __builtin_amdgcn_wmma_f32_16x16x4_f32(
                false, a[t], false, bB, (short)0, acc[2 * t], false, false);
            acc[2 * t + 1] = __builtin_amdgcn_wmma_f32_16x16x4_f32(
                false, a[t], false, bS, (short)0, acc[2 * t + 1], false, false);
        }
    }
#pragma unroll
    for (int t = 0; t < 4; ++t) {
        const int rbase = m0 + t * 16 + (lane >> 4) * 8;
#pragma unroll
        for (int v = 0; v < 8; ++v) {
            Hb[(size_t)(rbase + v) * NC + col] = acc[2 * t][v];
            Hs[(size_t)(rbase + v) * NC + col] = acc[2 * t + 1][v];
        }
    }
}

// ---------------------------------------------------------------------------
// 2) Per-node attention scalars: s1[i] = h_i . a1, s2[i] = h_i . a2
// ---------------------------------------------------------------------------
__global__ void k_node_dots(const float* __restrict__ H,
                            const float* __restrict__ a1,
                            const float* __restrict__ a2,
                            float* __restrict__ s1, float* __restrict__ s2) {
    int node = blockIdx.x * blockDim.y + threadIdx.y;
    if (node >= NTOT) return;
    int lane = threadIdx.x;
    const float* hr = H + (size_t)node * NC;
    float p1 = 0.f, p2 = 0.f;
#pragma unroll
    for (int t = 0; t < 4; ++t) {
        int j = lane + 32 * t;
        float h = hr[j];
        p1 += h * a1[j];
        p2 += h * a2[j];
    }
#pragma unroll
    for (int m = 16; m >= 1; m >>= 1) {
        p1 += __shfl_xor(p1, m, 32);
        p2 += __shfl_xor(p2, m, 32);
    }
    if (lane == 0) { s1[node] = p1; s2[node] = p2; }
}

// ---------------------------------------------------------------------------
// 3) In-degree (edge_index shared across batch/graphs -> computed once)
// ---------------------------------------------------------------------------
__global__ void k_deg(const int* __restrict__ EI, int E, int* __restrict__ deg) {
    int e = blockIdx.x * blockDim.x + threadIdx.x;
    if (e >= E) return;
    atomicAdd(&deg[EI[E + e]], 1);
}

// ---------------------------------------------------------------------------
// 4) Bend edges: score + pre-softmax alpha + score-sum per dst (for self loops)
// ---------------------------------------------------------------------------
__global__ void k_bend_edges(const int* __restrict__ EI, const float* __restrict__ attr,
                             const float* __restrict__ ssrc, const float* __restrict__ sdst,
                             const float* __restrict__ params,
                             float* __restrict__ alpha, float* __restrict__ esum) {
    int g = blockIdx.x * blockDim.x + threadIdx.x;
    if (g >= NB * EB) return;
    int b = g >> 16, e = g & (EB - 1);
    int src = EI[e], dst = EI[EB + e];
    const float* a4 = attr + (size_t)g * 4;
    float esc = a4[0] * params[0] + a4[1] * params[1] + a4[2] * params[2] + a4[3] * params[3] + params[4];
    float al = ssrc[b * NNODE + src] + sdst[b * NNODE + dst] + esc;
    alpha[g] = LEAKY(al, 0.2f);
    atomicAdd(&esum[b * NNODE + dst], esc);
}

// Section edge scores depend only on (batch, edge) -> B*ES of them
__global__ void k_sec_escore(const int* __restrict__ EI, const float* __restrict__ attr,
                             const float* __restrict__ params,
                             float* __restrict__ escore, float* __restrict__ esum) {
    int g = blockIdx.x * blockDim.x + threadIdx.x;
    if (g >= NB * ES) return;
    int b = g >> 12, e = g & (ES - 1);
    int dst = EI[ES + e];
    const float* a4 = attr + (size_t)g * 4;
    float esc = a4[0] * params[5] + a4[1] * params[6] + a4[2] * params[7] + a4[3] * params[8] + params[9];
    escore[g] = esc;
    atomicAdd(&esum[b * NN + dst], esc);
}

// Section alphas over all NG graphs
__global__ void k_sec_alpha(const int* __restrict__ EI, const float* __restrict__ escore,
                            const float* __restrict__ ssrc, const float* __restrict__ sdst,
                            float* __restrict__ alpha) {
    int g = blockIdx.x * blockDim.x + threadIdx.x;
    if (g >= NG * ES) return;
    int gg = g >> 12, e = g & (ES - 1);
    int b = gg >> 4;  // gg / NS
    int src = EI[e], dst = EI[ES + e];
    float al = ssrc[gg * NN + src] + sdst[gg * NN + dst] + escore[b * ES + e];
    alpha[g] = LEAKY(al, 0.2f);
}

// ---------------------------------------------------------------------------
// 5) Self-loop alpha + amax init (self loop always exists -> init, not -inf)
// ---------------------------------------------------------------------------
__global__ void k_self_init(const float* __restrict__ ssb, const float* __restrict__ sdb,
                            const float* __restrict__ sss, const float* __restrict__ sds,
                            const float* __restrict__ esum_b, const float* __restrict__ esum_s,
                            const int* __restrict__ deg_b, const int* __restrict__ deg_s,
                            float* __restrict__ self_b, float* __restrict__ self_s,
                            int* __restrict__ amax_b, int* __restrict__ amax_s) {
    int i = blockIdx.x * blockDim.x + threadIdx.x;
    if (i >= NTOT) return;
    int nodeN = i & (NNODE - 1);
    int b = i >> 12;         // i / NNODE
    int loc = i & (NN - 1);  // i % NN
    int db = deg_b[nodeN]; if (db < 1) db = 1;
    float sa = ssb[i] + sdb[i] + esum_b[i] / (float)db;
    sa = LEAKY(sa, 0.2f);
    self_b[i] = sa;
    amax_b[i] = ordf(sa);
    int ds = deg_s[loc]; if (ds < 1) ds = 1;
    float ss = sss[i] + sds[i] + esum_s[b * NN + loc] / (float)ds;
    ss = LEAKY(ss, 0.2f);
    self_s[i] = ss;
    amax_s[i] = ordf(ss);
}

// 6) Segment max over edges (monotonic int mapping -> atomicMax)
__global__ void k_amax_bend(const int* __restrict__ EI, const float* __restrict__ alpha,
                            int* __restrict__ amax) {
    int g = blockIdx.x * blockDim.x + threadIdx.x;
    if (g >= NB * EB) return;
    int b = g >> 16, e = g & (EB - 1);
    atomicMax(&amax[b * NNODE + EI[EB + e]], ordf(alpha[g]));
}
__global__ void k_amax_sec(const int* __restrict__ EI, const float* __restrict__ alpha,
                           int* __restrict__ amax) {
    int g = blockIdx.x * blockDim.x + threadIdx.x;
    if (g >= NG * ES) return;
    int gg = g >> 12, e = g & (ES - 1);
    atomicMax(&amax[gg * NN + EI[ES + e]], ordf(alpha[g]));
}

// 7) Denominator init with the self-loop term; selfw used in final fuse
__global__ void k_den_init(const float* __restrict__ self_b, const float* __restrict__ self_s,
                           const int* __restrict__ amax_b, const int* __restrict__ amax_s,
                           float* __restrict__ den_b, float* __restrict__ den_s,
                           float* __restrict__ sw_b, float* __restrict__ sw_s) {
    int i = blockIdx.x * blockDim.x + threadIdx.x;
    if (i >= NTOT) return;
    float eb = expf(self_b[i] - unordf(amax_b[i]));
    den_b[i] = eb; sw_b[i] = eb;
    float es = expf(self_s[i] - unordf(amax_s[i]));
    den_s[i] = es; sw_s[i] = es;
}

// 8) Scatter: wave per edge, lane = 4 channels (float4 gather, 4 atomics)
__global__ void k_scatter_bend(const int* __restrict__ EI, const float* __restrict__ alpha,
                               const int* __restrict__ amax, const float* __restrict__ H,
                               float* __restrict__ den, float* __restrict__ outu) {
    int w = blockIdx.x * blockDim.y + threadIdx.y;
    if (w >= NB * EB) return;
    int lane = threadIdx.x;
    int b = w >> 16, e = w & (EB - 1);
    int si = b * NNODE + EI[e];
    int di = b * NNODE + EI[EB + e];
    float ex = expf(alpha[w] - unordf(amax[di]));
    if (lane == 0) atomicAdd(&den[di], ex);
    const float4* hp = (const float4*)(H + (size_t)si * NC);
    float4 v = hp[lane];
    float* op = outu + (size_t)di * NC + lane * 4;
    atomicAdd(op + 0, ex * v.x);
    atomicAdd(op + 1, ex * v.y);
    atomicAdd(op + 2, ex * v.z);
    atomicAdd(op + 3, ex * v.w);
}
__global__ void k_scatter_sec(const int* __restrict__ EI, const float* __restrict__ alpha,
                              const int* __restrict__ amax, const float* __restrict__ H,
                              float* __restrict__ den, float* __restrict__ outu) {
    int w = blockIdx.x * blockDim.y + threadIdx.y;
    if (w >= NG * ES) return;
    int lane = threadIdx.x;
    int gg = w >> 12, e = w & (ES - 1);
    int si = gg * NN + EI[e];
    int di = gg * NN + EI[ES + e];
    float ex = expf(alpha[w] - unordf(amax[di]));
    if (lane == 0) atomicAdd(&den[di], ex);
    const float4* hp = (const float4*)(H + (size_t)si * NC);
    float4 v = hp[lane];
    float* op = outu + (size_t)di * NC + lane * 4;
    atomicAdd(op + 0, ex * v.x);
    atomicAdd(op + 1, ex * v.y);
    atomicAdd(op + 2, ex * v.z);
    atomicAdd(op + 3, ex * v.w);
}

// 9) Normalize, add self contribution + bias, leaky(0.01) on bend, fuse
__global__ void k_fuse(const float* __restrict__ Hb, const float* __restrict__ Hs,
                       const float* __restrict__ outu_b, const float* __restrict__ outu_s,
                       const float* __restrict__ sw_b, const float* __restrict__ sw_s,
                       const float* __restrict__ den_b, const float* __restrict__ den_s,
                       const float* __restrict__ bias_b, const float* __restrict__ bias_s,
                       const float* __restrict__ params, float* __restrict__ out) {
    size_t idx = (size_t)blockIdx.x * blockDim.x + threadIdx.x;
    if (idx >= (size_t)NTOT * NC) return;
    int i = (int)(idx >> 7), c = (int)(idx & (NC - 1));
    float ob = (outu_b[idx] + sw_b[i] * Hb[idx]) / den_b[i] + bias_b[c];
    ob = LEAKY(ob, 0.01f);
    float os = (outu_s[idx] + sw_s[i] * Hs[idx]) / den_s[i] + bias_s[c];
    out[idx] = params[10] * ob + params[11] * os;
}

// ---------------------------------------------------------------------------
extern "C" void kernel_launch(void* const* d_in, const int* in_sizes, int n_in,
                              void* d_out, int out_size, void* d_ws, size_t ws_size,
                              hipStream_t stream) {
    const float* x     = (const float*)d_in[0];
    const int*   sEI   = (const int*)d_in[1];
    const int*   bEI   = (const int*)d_in[2];
    const float* sAttr = (const float*)d_in[3];
    const float* bAttr = (const float*)d_in[4];
    const float* encW  = (const float*)d_in[5];
    const float* encB  = (const float*)d_in[6];
    const float* Wb    = (const float*)d_in[7];
    const float* asrcb = (const float*)d_in[8];
    const float* adstb = (const float*)d_in[9];
    const float* Web   = (const float*)d_in[10];
    const float* aeb   = (const float*)d_in[11];
    const float* biasb = (const float*)d_in[12];
    const float* Ws    = (const float*)d_in[13];
    const float* asrcs = (const float*)d_in[14];
    const float* adsts = (const float*)d_in[15];
    const float* Wes   = (const float*)d_in[16];
    const float* aes   = (const float*)d_in[17];
    const float* biass = (const float*)d_in[18];
    const float* fusew = (const float*)d_in[19];
    float* out = (float*)d_out;

    char* wsb = (char*)d_ws;
    size_t off = 0;
    auto alloc = [&](size_t bytes) -> char* {
        char* p = wsb + off;
        off += (bytes + 255) & ~(size_t)255;
        return p;
    };
    float* params  = (float*)alloc(16 * 4);
    float* Hb      = (float*)alloc((size_t)NTOT * NC * 4);
    float* Hs      = (float*)alloc((size_t)NTOT * NC * 4);
    float* ssb     = (float*)alloc(NTOT * 4);
    float* sdb     = (float*)alloc(NTOT * 4);
    float* sss     = (float*)alloc(NTOT * 4);
    float* sds     = (float*)alloc(NTOT * 4);
    int*   deg_b   = (int*)  alloc(NNODE * 4);
    int*   deg_s   = (int*)  alloc(NN * 4);
    float* esum_b  = (float*)alloc(NTOT * 4);
    float* esum_s  = (float*)alloc(NB * NN * 4);
    float* escore_s= (float*)alloc(NB * ES * 4);
    float* alpha_b = (float*)alloc((size_t)NB * EB * 4);
    float* alpha_s = (float*)alloc((size_t)NG * ES * 4);
    float* self_b  = (float*)alloc(NTOT * 4);
    float* self_s  = (float*)alloc(NTOT * 4);
    int*   amax_b  = (int*)  alloc(NTOT * 4);
    int*   amax_s  = (int*)  alloc(NTOT * 4);
    float* den_b   = (float*)alloc(NTOT * 4);
    float* den_s   = (float*)alloc(NTOT * 4);
    float* sw_b    = (float*)alloc(NTOT * 4);
    float* sw_s    = (float*)alloc(NTOT * 4);
    float* outu_b  = (float*)alloc((size_t)NTOT * NC * 4);
    float* outu_s  = (float*)alloc((size_t)NTOT * NC * 4);

    // zero the accumulators (harness poisons ws; must re-zero every call)
    hipMemsetAsync(deg_b, 0, NNODE * 4, stream);
    hipMemsetAsync(deg_s, 0, NN * 4, stream);
    hipMemsetAsync(esum_b, 0, NTOT * 4, stream);
    hipMemsetAsync(esum_s, 0, NB * NN * 4, stream);
    hipMemsetAsync(outu_b, 0, (size_t)NTOT * NC * 4, stream);
    hipMemsetAsync(outu_s, 0, (size_t)NTOT * NC * 4, stream);

    k_precompute<<<1, 32, 0, stream>>>(encW, encB, Web, aeb, Wes, aes, fusew, params);

    // WMMA GEMM: 512 M-groups (4 tiles each, 4 waves/block) x 8 N-tiles
    k_gemm_proj<<<dim3(128, 8), dim3(32, 4), 0, stream>>>(x, Wb, Ws, Hb, Hs);

    k_node_dots<<<NTOT / 8, dim3(32, 8), 0, stream>>>(Hb, asrcb, adstb, ssb, sdb);
    k_node_dots<<<NTOT / 8, dim3(32, 8), 0, stream>>>(Hs, asrcs, adsts, sss, sds);

    k_deg<<<EB / 256, 256, 0, stream>>>(bEI, EB, deg_b);
    k_deg<<<ES / 256, 256, 0, stream>>>(sEI, ES, deg_s);

    k_bend_edges<<<(NB * EB) / 256, 256, 0, stream>>>(bEI, bAttr, ssb, sdb, params, alpha_b, esum_b);
    k_sec_escore<<<(NB * ES) / 256, 256, 0, stream>>>(sEI, sAttr, params, escore_s, esum_s);
    k_sec_alpha<<<(NG * ES) / 256, 256, 0, stream>>>(sEI, escore_s, sss, sds, alpha_s);

    k_self_init<<<NTOT / 256, 256, 0, stream>>>(ssb, sdb, sss, sds, esum_b, esum_s,
                                                deg_b, deg_s, self_b, self_s, amax_b, amax_s);

    k_amax_bend<<<(NB * EB) / 256, 256, 0, stream>>>(bEI, alpha_b, amax_b);
    k_amax_sec<<<(NG * ES) / 256, 256, 0, stream>>>(sEI, alpha_s, amax_s);

    k_den_init<<<NTOT / 256, 256, 0, stream>>>(self_b, self_s, amax_b, amax_s,
                                               den_b, den_s, sw_b, sw_s);

    k_scatter_bend<<<(NB * EB) / 8, dim3(32, 8), 0, stream>>>(bEI, alpha_b, amax_b, Hb, den_b, outu_b);
    k_scatter_sec<<<(NG * ES) / 8, dim3(32, 8), 0, stream>>>(sEI, alpha_s, amax_s, Hs, den_s, outu_s);

    k_fuse<<<(NTOT * NC) / 256, 256, 0, stream>>>(Hb, Hs, outu_b, outu_s, sw_b, sw_s,
                                                  den_b, den_s, biasb, biass, params, out);
}